// AttentiveGRU_11158325035412
// MI455X (gfx1250) — compile-verified
//
#include <hip/hip_runtime.h>

typedef float v2f __attribute__((ext_vector_type(2)));
typedef float v8f __attribute__((ext_vector_type(8)));

#define HID 64
#define DIN 128

// V_WMMA_F32_16X16X4_F32: full-precision f32 matrix core op (CDNA5).
// 8-arg form: (neg_a, A, neg_b, B, c_mod, C, reuse_a, reuse_b)
static __device__ __forceinline__ v8f wmma4(v2f a, v2f b, v8f c) {
  return __builtin_amdgcn_wmma_f32_16x16x4_f32(false, a, false, b, (short)0, c,
                                               false, false);
}

__global__ __launch_bounds__(256) void zero_f32(float* __restrict__ p, int n) {
  int i = blockIdx.x * blockDim.x + threadIdx.x;
  int stride = gridDim.x * blockDim.x;
  for (; i < n; i += stride) p[i] = 0.0f;
}

// node_proj[N,64] = node_feats[N,128] @ W_node[64,128]^T
// Wave = one 16x16 output tile. Block = 8 waves = 32 rows x 64 cols.
__global__ __launch_bounds__(256) void node_proj_kernel(
    const float* __restrict__ X, const float* __restrict__ W,
    float* __restrict__ Y, int N) {
  const int lane = threadIdx.x & 31;
  const int wid  = threadIdx.x >> 5;
  const int lo  = lane & 15;
  const int hi2 = (lane >> 4) * 2;
  const int hi8 = (lane >> 4) * 8;
  const int ct  = wid & 3;                       // column tile (16 cols)
  const int nb  = blockIdx.x * 32 + (wid >> 2) * 16;
  if (nb >= N) return;
  const int colg = ct * 16 + lo;                 // 0..63
  const float* xrow = X + (size_t)(nb + lo) * DIN;
  const float* wrow = W + (size_t)colg * DIN;
  v8f acc = {};
#pragma unroll
  for (int k = 0; k < DIN; k += 4) {
    v2f a = *(const v2f*)(xrow + k + hi2);       // A[m][k..k+1], f32 16x4 layout
    v2f b = *(const v2f*)(wrow + k + hi2);       // B[k..k+1][n] = W[n][k..k+1]
    acc = wmma4(a, b, acc);
  }
#pragma unroll
  for (int r = 0; r < 8; ++r) {
    int row = nb + r + hi8;                      // C layout: M = r + 8*(lane>=16)
    Y[(size_t)row * HID + colg] = acc[r];
  }
}

// Fused edge pipeline: edge_proj GEMM + gather(node_proj[src]) + softmax(dim=hid)
// + node_hidden[src]*alpha + atomic scatter-add into h_new[dst].
// Wave = 16 edges x 64 hidden (4 WMMA accumulators).
__global__ __launch_bounds__(256) void edge_kernel(
    const float* __restrict__ EF, const float* __restrict__ We,
    const float* __restrict__ NP, const float* __restrict__ NH,
    const int* __restrict__ src, const int* __restrict__ dst,
    float* __restrict__ HNEW, int E) {
  const int lane = threadIdx.x & 31;
  const int wid  = threadIdx.x >> 5;
  const int lo  = lane & 15;
  const int hi2 = (lane >> 4) * 2;
  const int hi8 = (lane >> 4) * 8;
  const int eb  = (blockIdx.x * 8 + wid) * 16;
  if (eb >= E) return;
  const float* arow = EF + (size_t)(eb + lo) * DIN;
  const float* w0 = We + (size_t)(lo)      * DIN;
  const float* w1 = We + (size_t)(16 + lo) * DIN;
  const float* w2 = We + (size_t)(32 + lo) * DIN;
  const float* w3 = We + (size_t)(48 + lo) * DIN;
  v8f acc0 = {}, acc1 = {}, acc2 = {}, acc3 = {};
#pragma unroll
  for (int k = 0; k < DIN; k += 4) {
    v2f a = *(const v2f*)(arow + k + hi2);
    acc0 = wmma4(a, *(const v2f*)(w0 + k + hi2), acc0);
    acc1 = wmma4(a, *(const v2f*)(w1 + k + hi2), acc1);
    acc2 = wmma4(a, *(const v2f*)(w2 + k + hi2), acc2);
    acc3 = wmma4(a, *(const v2f*)(w3 + k + hi2), acc3);
  }
#pragma unroll
  for (int r = 0; r < 8; ++r) {
    const int e = eb + r + hi8;                  // edge id for this C row
    const int s = src[e];
    const int d = dst[e];
    const float* np = NP + (size_t)s * HID;
    const float* nh = NH + (size_t)s * HID;
    // edge_msg = edge_proj + node_proj[src]
    float v0 = acc0[r] + np[lo];
    float v1 = acc1[r] + np[16 + lo];
    float v2 = acc2[r] + np[32 + lo];
    float v3 = acc3[r] + np[48 + lo];
    // softmax over hidden dim: 64 values = 4 per lane across one 16-lane half
    float mx = fmaxf(fmaxf(v0, v1), fmaxf(v2, v3));
#pragma unroll
    for (int m = 1; m < 16; m <<= 1) mx = fmaxf(mx, __shfl_xor(mx, m, 32));
    float e0 = __expf(v0 - mx), e1 = __expf(v1 - mx);
    float e2 = __expf(v2 - mx), e3 = __expf(v3 - mx);
    float sum = e0 + e1 + e2 + e3;
#pragma unroll
    for (int m = 1; m < 16; m <<= 1) sum += __shfl_xor(sum, m, 32);
    const float inv = 1.0f / sum;
    float* out = HNEW + (size_t)d * HID;
    atomicAdd(out + lo,      nh[lo]      * e0 * inv);
    atomicAdd(out + 16 + lo, nh[16 + lo] * e1 * inv);
    atomicAdd(out + 32 + lo, nh[32 + lo] * e2 * inv);
    atomicAdd(out + 48 + lo, nh[48 + lo] * e3 * inv);
  }
}

// GRUCell: gi = h_new @ W_ih^T + b_ih ; gh = h @ W_hh^T + b_hh ; gated output.
// Wave = 16 nodes x 16 hidden cols, 6 WMMA accumulators (r/z/n x input/hidden).
__global__ __launch_bounds__(256) void gru_kernel(
    const float* __restrict__ HNEW, const float* __restrict__ NH,
    const float* __restrict__ Wih, const float* __restrict__ Whh,
    const float* __restrict__ bih, const float* __restrict__ bhh,
    float* __restrict__ OUT, int N) {
  const int lane = threadIdx.x & 31;
  const int wid  = threadIdx.x >> 5;
  const int lo  = lane & 15;
  const int hi2 = (lane >> 4) * 2;
  const int hi8 = (lane >> 4) * 8;
  const int ct  = wid & 3;
  const int nb  = blockIdx.x * 32 + (wid >> 2) * 16;
  if (nb >= N) return;
  const int colh = ct * 16 + lo;                 // 0..63 within H
  const float* xi  = HNEW + (size_t)(nb + lo) * HID;
  const float* xh  = NH   + (size_t)(nb + lo) * HID;
  const float* wir = Wih + (size_t)(colh)       * HID;
  const float* wiz = Wih + (size_t)(64  + colh) * HID;
  const float* win = Wih + (size_t)(128 + colh) * HID;
  const float* whr = Whh + (size_t)(colh)       * HID;
  const float* whz = Whh + (size_t)(64  + colh) * HID;
  const float* whn = Whh + (size_t)(128 + colh) * HID;
  v8f air = {}, aiz = {}, ain = {}, ahr = {}, ahz = {}, ahn = {};
#pragma unroll
  for (int k = 0; k < HID; k += 4) {
    v2f ai = *(const v2f*)(xi + k + hi2);
    v2f ah = *(const v2f*)(xh + k + hi2);
    air = wmma4(ai, *(const v2f*)(wir + k + hi2), air);
    aiz = wmma4(ai, *(const v2f*)(wiz + k + hi2), aiz);
    ain = wmma4(ai, *(const v2f*)(win + k + hi2), ain);
    ahr = wmma4(ah, *(const v2f*)(whr + k + hi2), ahr);
    ahz = wmma4(ah, *(const v2f*)(whz + k + hi2), ahz);
    ahn = wmma4(ah, *(const v2f*)(whn + k + hi2), ahn);
  }
  const float bir = bih[colh], biz = bih[64 + colh], bin = bih[128 + colh];
  const float bhr = bhh[colh], bhz = bhh[64 + colh], bhn = bhh[128 + colh];
#pragma unroll
  for (int r = 0; r < 8; ++r) {
    const int row = nb + r + hi8;
    float ir = air[r] + bir, iz = aiz[r] + biz, inn = ain[r] + bin;
    float hr = ahr[r] + bhr, hz = ahz[r] + bhz, hn  = ahn[r] + bhn;
    float rg = 1.0f / (1.0f + __expf(-(ir + hr)));
    float zg = 1.0f / (1.0f + __expf(-(iz + hz)));
    float ng = tanhf(inn + rg * hn);
    float hp = NH[(size_t)row * HID + colh];
    OUT[(size_t)row * HID + colh] = (1.0f - zg) * ng + zg * hp;
  }
}

extern "C" void kernel_launch(void* const* d_in, const int* in_sizes, int n_in,
                              void* d_out, int out_size, void* d_ws, size_t ws_size,
                              hipStream_t stream) {
  const float* node_feats  = (const float*)d_in[0];
  const float* edge_feats  = (const float*)d_in[1];
  const float* node_hidden = (const float*)d_in[2];
  const int*   src         = (const int*)d_in[3];
  const int*   dst         = (const int*)d_in[4];
  const float* W_edge      = (const float*)d_in[5];
  const float* W_node      = (const float*)d_in[6];
  const float* W_ih        = (const float*)d_in[7];
  const float* W_hh        = (const float*)d_in[8];
  const float* b_ih        = (const float*)d_in[9];
  const float* b_hh        = (const float*)d_in[10];
  float* out = (float*)d_out;

  const int N = in_sizes[2] / HID;   // 100000
  const int E = in_sizes[3];         // 1600000

  float* node_proj = (float*)d_ws;                       // N*64 f32
  float* h_new     = node_proj + (size_t)N * HID;        // N*64 f32

  zero_f32<<<512, 256, 0, stream>>>(h_new, N * HID);
  node_proj_kernel<<<(N + 31) / 32, 256, 0, stream>>>(node_feats, W_node,
                                                      node_proj, N);
  edge_kernel<<<(E + 127) / 128, 256, 0, stream>>>(edge_feats, W_edge,
                                                   node_proj, node_hidden,
                                                   src, dst, h_new, E);
  gru_kernel<<<(N + 31) / 32, 256, 0, stream>>>(h_new, node_hidden, W_ih, W_hh,
                                                b_ih, b_hh, out, N);
}